// ConvCapsules_44581760533196
// MI455X (gfx1250) — compile-verified
//
#include <hip/hip_runtime.h>
#include <hip/hip_bf16.h>

// ---------------- problem constants ----------------
#define BATCH 8
#define CSP 14          // CHILD_SPACE
#define CCAP 32         // CHILD_CAPS
#define PCAP 32         // PARENT_CAPS
#define PSP 6           // PARENT_SPACE
#define P2 36
#define KK 9
#define KCH 288         // KK*CCAP
#define NQ 288          // BATCH*P2 positions
#define EPSR 1e-9f
#define LOG2PI 1.8378770664093453f

// NOTE: votes / mean / i2v use PERMUTED pose-element order e' = c*4 + a
// (reference order is e = a*4 + c). All routing math is symmetric over e;
// only output_kernel permutes back. This makes the WMMA C-fragment store
// contiguous per lane (b128 stores, 256B-contiguous across lanes).

typedef float v2f __attribute__((ext_vector_type(2)));
typedef float v8f __attribute__((ext_vector_type(8)));

// number of parent windows covering child coordinate x (per-dimension)
__device__ __forceinline__ int cov_lo(int x) { int lo = (x - 1) / 2; return lo < 0 ? 0 : lo; }
__device__ __forceinline__ int cov_hi(int x) { int hi = x >> 1; return hi > 5 ? 5 : hi; }
__device__ __forceinline__ int cov_cnt(int x) {
    int lo = cov_lo(x), hi = cov_hi(x);
    return hi >= lo ? hi - lo + 1 : 0;
}

// flat offset into inputs[n][sr][sc][cap][17] for position q, child k
__device__ __forceinline__ size_t child_base(int q, int k) {
    int n = q / P2, p = q % P2;
    int r = p / PSP, c = p % PSP;
    int kk = k >> 5, cap = k & 31;
    int ii = kk / 3, jj = kk % 3;
    return ((((size_t)n * CSP + (r * 2 + ii)) * CSP + (c * 2 + jj)) * CCAP + cap) * 17;
}

// ---------------- weight transpose: wT[k][(o,c)][b] = w[k][o][b][c] ----------------
__global__ void wtrans_kernel(const float* __restrict__ w, float* __restrict__ wT) {
    int t = blockIdx.x * 256 + threadIdx.x;
    if (t >= KCH * 128 * 4) return;
    int k = t / 512, rem = t % 512;
    int col = rem >> 2, b = rem & 3;     // col = o*4 + c
    int o = col >> 2, c = col & 3;
    wT[t] = w[((size_t)(k * PCAP + o) * 4 + b) * 4 + c];
}

// ---------------- Phase 1: votes via V_WMMA_F32_16X16X4_F32 ----------------
// Per child k: GEMM  A[(q,a),b] (1152x4)  x  B[b,(o,c)] (4x128)  -> C (1152x128)
// One wave = one (k, 16-row M-tile); A fragment reused across 8 N-tiles.
__global__ __launch_bounds__(256) void votes_wmma_kernel(
    const float* __restrict__ in, const float* __restrict__ wT,
    float* __restrict__ votes) {
    int wave = threadIdx.x >> 5, lane = threadIdx.x & 31;
    int task = blockIdx.x * 8 + wave;      // 2592*8 == 288*72 exactly
    int k = task / 72, mt = task % 72;
    int half = lane >> 4, l15 = lane & 15;

    // A fragment: lane l15 holds row M = mt*16+l15; VGPR0/1 = K {0,1} (half=0) or {2,3} (half=1)
    int Rm = mt * 16 + l15;
    int q = Rm >> 2, a = Rm & 3;
    size_t base = child_base(q, k);
    v2f afr;
    afr.x = in[base + a * 4 + half * 2 + 0];
    afr.y = in[base + a * 4 + half * 2 + 1];

    int qq0 = mt * 4 + half * 2;           // C rows v=0..3 -> qq0, v=4..7 -> qq0+1
    for (int nt = 0; nt < 8; ++nt) {
        int col = nt * 16 + l15;           // N = o*4 + c
        int oo = col >> 2, cc2 = col & 3;
        // B fragment: single aligned b64 load from transposed weights
        v2f bfr = *(const v2f*)(wT + ((size_t)k * 128 + col) * 4 + half * 2);
        v8f acc = {};
        acc = __builtin_amdgcn_wmma_f32_16x16x4_f32(
            false, afr, false, bfr, (short)0, acc, false, false);
        // permuted layout: votes[q][k][o][c*4 + a] -> lane's 4 a-values contiguous
        float4 lo4, hi4;
        lo4.x = acc[0]; lo4.y = acc[1]; lo4.z = acc[2]; lo4.w = acc[3];
        hi4.x = acc[4]; hi4.y = acc[5]; hi4.z = acc[6]; hi4.w = acc[7];
        size_t b0 = (((size_t)qq0 * KCH + k) * PCAP + oo) * 16 + cc2 * 4;
        size_t b1 = (((size_t)(qq0 + 1) * KCH + k) * PCAP + oo) * 16 + cc2 * 4;
        *(float4*)(votes + b0) = lo4;
        *(float4*)(votes + b1) = hi4;
    }
}

// ---------------- tiled activations gather ----------------
__global__ void act_gather_kernel(const float* __restrict__ in, float* __restrict__ act_t) {
    int t = blockIdx.x * 256 + threadIdx.x;
    if (t >= NQ * KCH) return;
    int q = t / KCH, k = t % KCH;
    act_t[t] = in[child_base(q, k) + 16];
}

// ---------------- M-step (fused rr normalization + moments + activations) ----------------
// block = 512 threads = (o,e') pairs, one block per position q. Single pass over votes.
__global__ __launch_bounds__(512) void mstep_kernel(
    const float* __restrict__ votes, const float* __restrict__ act_t,
    const float* __restrict__ e_buf, const float* __restrict__ denom,
    const float* __restrict__ beta_v, const float* __restrict__ beta_a,
    float* __restrict__ mean_g, float* __restrict__ i2v_g,
    float* __restrict__ lnA_g, float* __restrict__ actj_g,
    float inv_temp, int init_rr) {
    __shared__ float act_s[KCH];
    __shared__ float rr_s[KCH * PCAP];
    __shared__ float cost_s[PCAP];
    __shared__ float L_s[PCAP];

    int q = blockIdx.x, t = threadIdx.x;
    int n = q / P2, p = q % P2;
    int pr = p / PSP, pc = p % PSP;

    for (int i = t; i < KCH; i += 512) act_s[i] = act_t[(size_t)q * KCH + i];
    for (int i = t; i < KCH * PCAP; i += 512) {
        int k = i >> 5, o = i & 31;
        int kk = k >> 5, cap = k & 31;
        int ii = kk / 3, jj = kk % 3;
        float val;
        if (init_rr) {
            int ppc = cov_cnt(pr * 2 + ii) * cov_cnt(pc * 2 + jj);
            val = 1.0f / ((float)ppc * (float)PCAP);
        } else {                            // rr = e / (denom + eps), fused from E-step
            int s = (pr * 2 + ii) * CSP + (pc * 2 + jj);
            val = e_buf[((size_t)q * KCH + k) * PCAP + o] /
                  (denom[((size_t)n * CSP * CSP + s) * CCAP + cap] + EPSR);
        }
        rr_s[i] = val;
    }
    __syncthreads();

    int o = t >> 4;
    float S = 0.f, Sv = 0.f, Svv = 0.f;
    const float* vb = votes + (size_t)q * KCH * 512 + t;   // coalesced: 512 contiguous per k
    for (int k = 0; k < KCH; ++k) {
        float v = vb[(size_t)k * 512];
        float wr = rr_s[(k << 5) | o] * act_s[k];
        S += wr; Sv += wr * v; Svv += wr * v * v;
    }
    float Se = S + EPSR;
    float mean = Sv / Se;
    float var = (Svv - 2.f * mean * Sv + mean * mean * S) / Se + EPSR;
    mean_g[(size_t)q * 512 + t] = mean;
    i2v_g[(size_t)q * 512 + t] = 0.5f / var;

    float L = logf(var);                    // reduce Sum_e log(var) within 16-lane e-group
    for (int msk = 1; msk < 16; msk <<= 1) L += __shfl_xor(L, msk, 16);
    if ((t & 15) == 0) {
        cost_s[o] = (16.0f * beta_v[o] + 0.5f * L) * S;
        L_s[o] = L;
    }
    __syncthreads();
    if (t < PCAP) {                         // wave 0: cross-parent stats + sigmoid
        float cst = cost_s[t];
        float sum = cst;
        for (int msk = 1; msk < 32; msk <<= 1) sum += __shfl_xor(sum, msk, 32);
        float cmean = sum * (1.0f / 32.0f);
        float d2 = (cst - cmean) * (cst - cmean);
        for (int msk = 1; msk < 32; msk <<= 1) d2 += __shfl_xor(d2, msk, 32);
        float stdv = sqrtf(d2 * (1.0f / 32.0f));
        float x = inv_temp * (beta_a[t] + (cmean - cst) / (stdv + EPSR));
        float aj = 1.0f / (1.0f + expf(-x));
        actj_g[q * PCAP + t] = aj;
        lnA_g[q * PCAP + t] = logf(aj + EPSR) - 0.5f * (16.0f * LOG2PI + L_s[t]);
    }
}

// ---------------- E-step part A: zz and per-(q,k) max over parents ----------------
__global__ __launch_bounds__(256) void estep_zz_kernel(
    const float* __restrict__ votes, const float* __restrict__ mean_g,
    const float* __restrict__ i2v_g, const float* __restrict__ lnA_g,
    float* __restrict__ zz, float* __restrict__ zmax) {
    int q = blockIdx.x;
    int wave = threadIdx.x >> 5, lane = threadIdx.x & 31;  // lane = parent o
    float mu[16], iv[16];
    const float4* mp = (const float4*)(mean_g + (size_t)(q * PCAP + lane) * 16);
    const float4* ip = (const float4*)(i2v_g + (size_t)(q * PCAP + lane) * 16);
#pragma unroll
    for (int i = 0; i < 4; ++i) {
        float4 mv = mp[i]; mu[4*i]=mv.x; mu[4*i+1]=mv.y; mu[4*i+2]=mv.z; mu[4*i+3]=mv.w;
        float4 vv = ip[i]; iv[4*i]=vv.x; iv[4*i+1]=vv.y; iv[4*i+2]=vv.z; iv[4*i+3]=vv.w;
    }
    float lnA = lnA_g[q * PCAP + lane];
    for (int it = 0; it < 36; ++it) {
        int k = it * 8 + wave;
        const float4* vp = (const float4*)(votes + ((size_t)(q * KCH + k) * PCAP + lane) * 16);
        float s = 0.f;
#pragma unroll
        for (int i = 0; i < 4; ++i) {
            float4 v = vp[i];
            float d0 = v.x - mu[4*i],   d1 = v.y - mu[4*i+1];
            float d2 = v.z - mu[4*i+2], d3 = v.w - mu[4*i+3];
            s += d0*d0*iv[4*i] + d1*d1*iv[4*i+1] + d2*d2*iv[4*i+2] + d3*d3*iv[4*i+3];
        }
        float z = lnA - s;
        zz[((size_t)q * KCH + k) * PCAP + lane] = z;
        float zm = z;
        for (int msk = 1; msk < 32; msk <<= 1) zm = fmaxf(zm, __shfl_xor(zm, msk, 32));
        if (lane == 0) zmax[(size_t)q * KCH + k] = zm;
    }
}

// ---------------- E-step part B: per-child-cell max (gather over <=4 windows) ----------------
__global__ void cmax_kernel(const float* __restrict__ zmax, float* __restrict__ cmax) {
    int t = blockIdx.x * 256 + threadIdx.x;
    if (t >= BATCH * CSP * CSP * CCAP) return;
    int cap = t & 31;
    int s = (t >> 5) % (CSP * CSP);
    int n = t / (CSP * CSP * CCAP);
    int sr = s / CSP, sc = s % CSP;
    float m = -1e30f;
    for (int r = cov_lo(sr); r <= cov_hi(sr); ++r)
        for (int c = cov_lo(sc); c <= cov_hi(sc); ++c) {
            int kk = (sr - 2 * r) * 3 + (sc - 2 * c);
            m = fmaxf(m, zmax[(size_t)(n * P2 + r * PSP + c) * KCH + kk * CCAP + cap]);
        }
    cmax[t] = m;
}

// ---------------- E-step part C: e = exp(zz - cmax) in-place, row sums ----------------
__global__ __launch_bounds__(256) void estep_e_kernel(
    const float* __restrict__ cmax, float* __restrict__ zz_e, float* __restrict__ se) {
    int q = blockIdx.x;
    int wave = threadIdx.x >> 5, lane = threadIdx.x & 31;
    int n = q / P2, p = q % P2;
    int pr = p / PSP, pc = p % PSP;
    for (int it = 0; it < 36; ++it) {
        int k = it * 8 + wave;
        int kk = k >> 5, cap = k & 31;
        int s = (pr * 2 + kk / 3) * CSP + (pc * 2 + kk % 3);
        float cm = cmax[((size_t)n * CSP * CSP + s) * CCAP + cap];
        size_t idx = ((size_t)q * KCH + k) * PCAP + lane;
        float ev = expf(zz_e[idx] - cm);
        zz_e[idx] = ev;
        float sum = ev;
        for (int msk = 1; msk < 32; msk <<= 1) sum += __shfl_xor(sum, msk, 32);
        if (lane == 0) se[(size_t)q * KCH + k] = sum;
    }
}

// ---------------- E-step part D: per-child-cell denominator (gather-add) ----------------
__global__ void denom_kernel(const float* __restrict__ se, float* __restrict__ denom) {
    int t = blockIdx.x * 256 + threadIdx.x;
    if (t >= BATCH * CSP * CSP * CCAP) return;
    int cap = t & 31;
    int s = (t >> 5) % (CSP * CSP);
    int n = t / (CSP * CSP * CCAP);
    int sr = s / CSP, sc = s % CSP;
    float d = 0.f;
    for (int r = cov_lo(sr); r <= cov_hi(sr); ++r)
        for (int c = cov_lo(sc); c <= cov_hi(sc); ++c) {
            int kk = (sr - 2 * r) * 3 + (sc - 2 * c);
            d += se[(size_t)(n * P2 + r * PSP + c) * KCH + kk * CCAP + cap];
        }
    denom[t] = d;
}

// ---------------- final output: [act_j, pose(16)] per (q,o); un-permute e ----------------
__global__ void output_kernel(const float* __restrict__ actj,
                              const float* __restrict__ mean_g, float* __restrict__ out) {
    int t = blockIdx.x * 256 + threadIdx.x;
    if (t >= NQ * PCAP) return;
    float* dst = out + (size_t)t * 17;
    dst[0] = actj[t];
    const float* mg = mean_g + (size_t)t * 16;   // permuted: mg[c*4+a]
#pragma unroll
    for (int a = 0; a < 4; ++a)
#pragma unroll
        for (int c = 0; c < 4; ++c)
            dst[1 + a * 4 + c] = mg[c * 4 + a];
}

extern "C" void kernel_launch(void* const* d_in, const int* in_sizes, int n_in,
                              void* d_out, int out_size, void* d_ws, size_t ws_size,
                              hipStream_t stream) {
    const float* in     = (const float*)d_in[0];
    const float* w      = (const float*)d_in[1];
    const float* beta_v = (const float*)d_in[2];
    const float* beta_a = (const float*)d_in[3];

    float* ws = (float*)d_ws;
    size_t off = 0;
    float* votes  = ws + off; off += (size_t)NQ * KCH * PCAP * 16;  // ~170MB, L2-resident
    float* mean_g = ws + off; off += (size_t)NQ * PCAP * 16;
    float* i2v_g  = ws + off; off += (size_t)NQ * PCAP * 16;
    float* lnA    = ws + off; off += (size_t)NQ * PCAP;
    float* actj   = ws + off; off += (size_t)NQ * PCAP;
    float* zz     = ws + off; off += (size_t)NQ * KCH * PCAP;       // reused as e-buffer in place
    float* zmax   = ws + off; off += (size_t)NQ * KCH;
    float* se     = ws + off; off += (size_t)NQ * KCH;
    float* cmaxb  = ws + off; off += (size_t)BATCH * CSP * CSP * CCAP;
    float* denomb = ws + off; off += (size_t)BATCH * CSP * CSP * CCAP;
    float* act_t  = ws + off; off += (size_t)NQ * KCH;
    float* wT     = ws + off; off += (size_t)KCH * 128 * 4;

    wtrans_kernel<<<(KCH * 128 * 4 + 255) / 256, 256, 0, stream>>>(w, wT);
    votes_wmma_kernel<<<2592, 256, 0, stream>>>(in, wT, votes);
    act_gather_kernel<<<(NQ * KCH + 255) / 256, 256, 0, stream>>>(in, act_t);

    const float inv_temp[3] = {0.0005f, 0.000975f, 0.00142625f};  // lambda*(1-0.95^(it+1))
    for (int it = 0; it < 3; ++it) {
        mstep_kernel<<<NQ, 512, 0, stream>>>(votes, act_t, zz, denomb, beta_v, beta_a,
                                             mean_g, i2v_g, lnA, actj,
                                             inv_temp[it], it == 0 ? 1 : 0);
        if (it < 2) {
            estep_zz_kernel<<<NQ, 256, 0, stream>>>(votes, mean_g, i2v_g, lnA, zz, zmax);
            cmax_kernel<<<(BATCH * CSP * CSP * CCAP + 255) / 256, 256, 0, stream>>>(zmax, cmaxb);
            estep_e_kernel<<<NQ, 256, 0, stream>>>(cmaxb, zz, se);
            denom_kernel<<<(BATCH * CSP * CSP * CCAP + 255) / 256, 256, 0, stream>>>(se, denomb);
        }
    }
    output_kernel<<<(NQ * PCAP + 255) / 256, 256, 0, stream>>>(actj, mean_g, (float*)d_out);
}